// MaxPool2d_30073361006681
// MI455X (gfx1250) — compile-verified
//
#include <hip/hip_runtime.h>

namespace {
constexpr int H = 224, W = 224;
constexpr int OH = 111, OW = 111;          // (224-3)/2+1
constexpr int ROWS = 16;                   // output rows per workgroup
constexpr int IN_ROWS = 2 * ROWS + 1;      // 33 input rows staged in LDS
constexpr int THREADS = 256;               // 8 x wave32
constexpr int BLOCKS_PER_IMG = (OH + ROWS - 1) / ROWS;  // 7
}

__global__ __launch_bounds__(THREADS) void maxpool2d_k3s2_async(
    const float* __restrict__ in, float* __restrict__ out, int n_imgs) {
  // 33 rows x 224 floats = 29568 B of LDS (of 320 KB/WGP)
  __shared__ __align__(16) float tile[IN_ROWS * W];

  const int tid = threadIdx.x;
  const int bid = blockIdx.x;
  const int img = bid / BLOCKS_PER_IMG;
  if (img >= n_imgs) return;
  const int rb = bid - img * BLOCKS_PER_IMG;

  const int y0    = rb * ROWS;                           // first output row
  const int nrows = (OH - y0 < ROWS) ? (OH - y0) : ROWS; // 16 (15 for last strip)
  const int n_in  = 2 * nrows + 1;                       // staged input rows

  // Input rows y0*2 .. y0*2 + n_in - 1 are contiguous in memory.
  const float* src = in + (size_t)img * (H * W) + (size_t)(y0 * 2) * W;

  // ---- Stage input rows into LDS with CDNA5 async global->LDS copies.
  // local->flat addrspacecast puts the LDS byte offset in the low 32 bits.
  const unsigned lds_base = (unsigned)(unsigned long long)&tile[0];
  const int nvec = n_in * (W / 4);  // float4 chunks; 224 % 4 == 0
  for (int i = tid; i < nvec; i += THREADS) {
    unsigned lds_addr = lds_base + (unsigned)i * 16u;
    unsigned goff     = (unsigned)i * 16u;  // byte offset (GVS: saddr + vaddr)
    asm volatile("global_load_async_to_lds_b128 %0, %1, %2"
                 :
                 : "v"(lds_addr), "v"(goff), "s"(src)
                 : "memory");
  }
  asm volatile("s_wait_asynccnt 0" ::: "memory");
  __syncthreads();

  // ---- Max over 3x3 windows from LDS (conflict-free: bank = 2x mod 64).
  float* dst = out + (size_t)img * (OH * OW) + (size_t)y0 * OW;
  const int nout = nrows * OW;
  for (int o = tid; o < nout; o += THREADS) {
    const int yy = o / OW;          // local output row
    const int x  = o - yy * OW;     // output col
    const float* r0 = &tile[(2 * yy) * W + 2 * x];
    const float* r1 = r0 + W;
    const float* r2 = r1 + W;
    float m0 = fmaxf(fmaxf(r0[0], r0[1]), r0[2]);
    float m1 = fmaxf(fmaxf(r1[0], r1[1]), r1[2]);
    float m2 = fmaxf(fmaxf(r2[0], r2[1]), r2[2]);
    dst[(size_t)yy * OW + x] = fmaxf(fmaxf(m0, m1), m2);
  }
}

extern "C" void kernel_launch(void* const* d_in, const int* in_sizes, int n_in,
                              void* d_out, int out_size, void* d_ws, size_t ws_size,
                              hipStream_t stream) {
  (void)n_in; (void)out_size; (void)d_ws; (void)ws_size;
  const float* x = (const float*)d_in[0];
  float* out = (float*)d_out;
  const int n_imgs = in_sizes[0] / (H * W);  // 32 * 64 = 2048 independent planes
  const int grid = n_imgs * BLOCKS_PER_IMG;  // 14336 workgroups
  maxpool2d_k3s2_async<<<grid, THREADS, 0, stream>>>(x, out, n_imgs);
}